// HenonLayer_50337016709335
// MI455X (gfx1250) — compile-verified
//
#include <hip/hip_runtime.h>

typedef __attribute__((ext_vector_type(2))) float v2f;
typedef __attribute__((ext_vector_type(8))) float v8f;

#define NI 64
#define NTILE 4  // four 16-wide tiles cover the 64 hidden units

// gfx1250 hardware tanh (new transcendental on CDNA5). Fallback ladder:
// clang builtin -> inline asm.
__device__ __forceinline__ float fast_tanh(float x) {
#if __has_builtin(__builtin_amdgcn_tanhf)
  return __builtin_amdgcn_tanhf(x);
#else
  float r;
  asm volatile("v_tanh_f32 %0, %1" : "=v"(r) : "v"(x));
  return r;
#endif
}

__global__ __launch_bounds__(256) void henon_wmma_kernel(
    const float* __restrict__ z, const float* __restrict__ W_in,
    const float* __restrict__ W_out, const float* __restrict__ b_in,
    const float* __restrict__ eta, float* __restrict__ out, int nbatch) {
  // Per-wave transpose buffer: Gt[wave][j(hidden)][m(batch)], 4KB per wave.
  __shared__ float Gt[8][NI][16];

  const int tid   = threadIdx.x;
  const int lane  = tid & 31;
  const int wslot = tid >> 5;
  const int hilo  = lane >> 4;   // which 16-lane half
  const int l16   = lane & 15;
  float* gbase = &Gt[wslot][0][0];

  // ---- wave-invariant constant fragments (VGPR resident) ----
  // Matmul1 B fragment per hidden tile: rows K0=Win0, K1=Win1, K2=b_in, K3=0.
  // 32-bit B layout: vgpr0 = {K0 | K2}, vgpr1 = {K1 | K3} across lane halves.
  float b1x[NTILE], b1y[NTILE], wv[NTILE];
#pragma unroll
  for (int T = 0; T < NTILE; ++T) {
    int j = T * 16 + l16;
    b1x[T] = hilo ? b_in[j] : W_in[0 * NI + j];
    b1y[T] = hilo ? 0.0f    : W_in[1 * NI + j];
    wv[T]  = W_out[j];
  }
  // Matmul2 A fragments: A = W_in as (M=2 x K=64), chunked K=4.
  // 32-bit A layout: vgpr0 = {K=4c | K=4c+2}, vgpr1 = {K=4c+1 | K=4c+3},
  // M = lane%16 (rows 2..15 zero).
  float a2x[16], a2y[16];
#pragma unroll
  for (int c = 0; c < 16; ++c) {
    int k0 = 4 * c + 2 * hilo;
    a2x[c] = (l16 < 2) ? W_in[l16 * NI + k0]     : 0.0f;
    a2y[c] = (l16 < 2) ? W_in[l16 * NI + k0 + 1] : 0.0f;
  }
  const float eta0 = eta[0], eta1 = eta[1];

  const int ntiles = (nbatch + 15) >> 4;
  const int waveG  = (int)(blockIdx.x * blockDim.x + tid) >> 5;
  const int nwaves = (int)(gridDim.x * blockDim.x) >> 5;
  const float4* z4 = (const float4*)z;
  float4* o4 = (float4*)out;

  for (int tile = waveG; tile < ntiles; tile += nwaves) {
    const int b = tile * 16 + l16;
    const bool act = (hilo == 0) && (b < nbatch);
    float4 zv = make_float4(0.f, 0.f, 0.f, 0.f);
    if (act) zv = z4[b];
    {  // prefetch next tile's batch rows (global_prefetch_b8)
      int nt = tile + nwaves;
      if (nt < ntiles) __builtin_prefetch(&z4[nt * 16 + l16], 0, 1);
    }
    float X0 = zv.x, X1 = zv.y, Y0 = zv.z, Y1 = zv.w;

#pragma unroll
    for (int it = 0; it < 4; ++it) {
      // --- matmul1: H = [Y0 Y1 1 0] x [Win0; Win1; b; 0]  (bias via K) ---
      v2f a1;
      a1.x = hilo ? 1.0f : Y0;  // lanes 16-31 carry K=2 (=1.0) per A layout
      a1.y = hilo ? 0.0f : Y1;  // lanes 16-31 carry K=3 (=0.0)
      v8f h[NTILE];
#pragma unroll
      for (int T = 0; T < NTILE; ++T) {
        v2f b1; b1.x = b1x[T]; b1.y = b1y[T];
        v8f cz = {};
        h[T] = __builtin_amdgcn_wmma_f32_16x16x4_f32(false, a1, false, b1,
                                                     (short)0, cz, false, false);
      }
      // --- elementwise: G = (1 - tanh(H)^2) * W_out, store transposed ---
#pragma unroll
      for (int T = 0; T < NTILE; ++T) {
        float g[8];
#pragma unroll
        for (int r = 0; r < 8; ++r) {
          float t  = fast_tanh(h[T][r]);
          float tw = t * wv[T];
          g[r] = __builtin_fmaf(-tw, t, wv[T]);
        }
        // C layout: lane holds column j = T*16+l16, rows m = 8*hilo + r.
        // Write into Gt[j][m]: rows are contiguous -> two b128 stores.
        int addr = (T * 16 + l16) * 16 + 8 * hilo;
        *(float4*)(gbase + addr)     = make_float4(g[0], g[1], g[2], g[3]);
        *(float4*)(gbase + addr + 4) = make_float4(g[4], g[5], g[6], g[7]);
      }
      // --- matmul2: gradT = W_in (2x64) x G^T (64x16), K chunks of 4 ---
      v8f acc = {};
#pragma unroll
      for (int c = 0; c < 16; ++c) {
        int roff = (4 * c + 2 * hilo) * 16 + l16;
        v2f bf;
        bf.x = gbase[roff];        // row K=4c(+2), N = l16
        bf.y = gbase[roff + 16];   // row K=4c+1(+3)
        v2f af; af.x = a2x[c]; af.y = a2y[c];
        acc = __builtin_amdgcn_wmma_f32_16x16x4_f32(false, af, false, bf,
                                                    (short)0, acc, false, false);
      }
      // C rows 0/1 (lanes 0-15) = grad0/grad1 per batch element.
      float g0 = acc[0], g1 = acc[1];
      float nX0 = Y0 + eta0;
      float nX1 = Y1 + eta1;
      float nY0 = g0 - X0;   // eps = 1.0
      float nY1 = g1 - X1;
      X0 = nX0; X1 = nX1; Y0 = nY0; Y1 = nY1;
    }
    if (act) o4[b] = make_float4(X0, X1, Y0, Y1);
  }
}

extern "C" void kernel_launch(void* const* d_in, const int* in_sizes, int n_in,
                              void* d_out, int out_size, void* d_ws, size_t ws_size,
                              hipStream_t stream) {
  const float* z     = (const float*)d_in[0];
  const float* W_in  = (const float*)d_in[1];
  const float* W_out = (const float*)d_in[2];
  const float* b_in  = (const float*)d_in[3];
  const float* eta   = (const float*)d_in[4];
  float* out = (float*)d_out;
  int nbatch = in_sizes[0] / 4;  // z is (BATCH, 4) f32

  dim3 block(256);
  dim3 grid(2048);  // 16384 waves, grid-stride over 125k batch tiles
  henon_wmma_kernel<<<grid, block, 0, stream>>>(z, W_in, W_out, b_in, eta,
                                                out, nbatch);
}